// SparseIF_7284264534724
// MI455X (gfx1250) — compile-verified
//
#include <hip/hip_runtime.h>
#include <cstdint>
#include <cstddef>

// Integrate-and-fire, soft reset, T=64, fp32, shape [T, B*D] with B*D = 262144.
// Memory-bound streaming scan: async global->LDS 16-deep pipeline per wave to
// build ~16MB of in-flight loads (BW*latency product for 23.3 TB/s HBM).

typedef float f4 __attribute__((ext_vector_type(4)));
typedef int   v4i __attribute__((vector_size(16)));   // b128 payload type

#define AS1 __attribute__((address_space(1)))
#define AS3 __attribute__((address_space(3)))

#define IF_TSTEPS 64
#define IF_BLOCK  256
#define IF_DEPTH  16   // ring-buffer depth (timesteps in flight per wave)

#if defined(__HIP_DEVICE_COMPILE__) && defined(__gfx1250__) && \
    __has_builtin(__builtin_amdgcn_global_load_async_to_lds_b128)
#define IF_ASYNC_PATH 1
#else
#define IF_ASYNC_PATH 0
#endif

#if defined(__HIP_DEVICE_COMPILE__) && defined(__gfx1250__)
#if __has_builtin(__builtin_amdgcn_s_wait_asynccnt)
#define WAITA(n) __builtin_amdgcn_s_wait_asynccnt(n)
#else
#define WAITA(n) asm volatile("s_wait_asynccnt %0" ::"i"(n) : "memory")
#endif
#else
#define WAITA(n)
#endif

__device__ __forceinline__ void async_b128(const f4* g, f4* l) {
#if IF_ASYNC_PATH
  __builtin_amdgcn_global_load_async_to_lds_b128(
      (AS1 v4i*)(uintptr_t)g,
      (AS3 v4i*)(uint32_t)(uintptr_t)l,
      /*offset=*/0, /*cpol=*/0);
#else
  (void)g; (void)l;
#endif
}

__device__ __forceinline__ void if_step(f4& v, const f4 c, f4& s) {
#pragma unroll
  for (int k = 0; k < 4; ++k) {
    float vt = v[k] + c[k];
    float sp = (vt >= 1.0f) ? 1.0f : 0.0f;  // V_THRESHOLD = 1.0
    s[k] = sp;
    v[k] = vt - sp;                         // soft reset: v -= spike * th
  }
}

__global__ __launch_bounds__(IF_BLOCK) void if_soft_reset_kernel(
    const f4* __restrict__ cur, f4* __restrict__ out, int n4) {
  const int tid = threadIdx.x;
  const int gid = blockIdx.x * IF_BLOCK + tid;
  if (gid >= n4) return;

  const size_t stride = (size_t)n4;  // f4 elements per timestep slice
  f4 v = {0.f, 0.f, 0.f, 0.f};

#if IF_ASYNC_PATH
  // 16-deep ring in LDS: 256 lanes * 16 B * 16 slots = 64 KB per block.
  __shared__ f4 buf[IF_DEPTH][IF_BLOCK];
  const f4* gp = cur + gid;

  // Prologue: issue loads for slices 0 .. DEPTH-2 (15 outstanding).
#pragma unroll
  for (int j = 0; j < IF_DEPTH - 1; ++j)
    async_b128(gp + (size_t)j * stride, &buf[j][tid]);

#define IF_STEP_AT(t)                                    \
  do {                                                   \
    f4 c = buf[(t) & (IF_DEPTH - 1)][tid];               \
    f4 s;                                                \
    if_step(v, c, s);                                    \
    out[(size_t)(t) * stride + gid] = s;                 \
  } while (0)

  // Steady state: issue slice t+15 (slot (t-1)&15, already consumed at t-1),
  // then wait until <=15 outstanding => slice t has landed (in-order done).
#pragma unroll 1
  for (int t = 0; t < IF_TSTEPS - (IF_DEPTH - 1); ++t) {  // t = 0..48
    async_b128(gp + (size_t)(t + IF_DEPTH - 1) * stride,
               &buf[(t + IF_DEPTH - 1) & (IF_DEPTH - 1)][tid]);
    WAITA(IF_DEPTH - 1);
    IF_STEP_AT(t);
  }
  // Tail: all 64 loads issued; drain with exact constant waits.
  WAITA(14); IF_STEP_AT(49);
  WAITA(13); IF_STEP_AT(50);
  WAITA(12); IF_STEP_AT(51);
  WAITA(11); IF_STEP_AT(52);
  WAITA(10); IF_STEP_AT(53);
  WAITA(9);  IF_STEP_AT(54);
  WAITA(8);  IF_STEP_AT(55);
  WAITA(7);  IF_STEP_AT(56);
  WAITA(6);  IF_STEP_AT(57);
  WAITA(5);  IF_STEP_AT(58);
  WAITA(4);  IF_STEP_AT(59);
  WAITA(3);  IF_STEP_AT(60);
  WAITA(2);  IF_STEP_AT(61);
  WAITA(1);  IF_STEP_AT(62);
  WAITA(0);  IF_STEP_AT(63);
#undef IF_STEP_AT

#else
  // Fallback (builtins unavailable): register-resident 8-deep prefetch.
  f4 p[8];
#pragma unroll
  for (int j = 0; j < 8; ++j) p[j] = cur[(size_t)j * stride + gid];
#pragma unroll 1
  for (int base = 0; base < IF_TSTEPS; base += 8) {
    f4 q[8];
    const bool more = (base + 8) < IF_TSTEPS;
#pragma unroll
    for (int j = 0; j < 8; ++j)
      if (more) q[j] = cur[(size_t)(base + 8 + j) * stride + gid];
#pragma unroll
    for (int j = 0; j < 8; ++j) {
      f4 s;
      if_step(v, p[j], s);
      out[(size_t)(base + j) * stride + gid] = s;
    }
#pragma unroll
    for (int j = 0; j < 8; ++j)
      if (more) p[j] = q[j];
  }
#endif
}

extern "C" void kernel_launch(void* const* d_in, const int* in_sizes, int n_in,
                              void* d_out, int out_size, void* d_ws, size_t ws_size,
                              hipStream_t stream) {
  (void)n_in; (void)out_size; (void)d_ws; (void)ws_size;
  const float* cur = (const float*)d_in[0];
  float* out = (float*)d_out;

  const int total = in_sizes[0];          // T * B * D = 16777216
  const int slice = total / IF_TSTEPS;    // B * D    = 262144
  const int n4 = slice / 4;               //          = 65536 float4 lanes

  dim3 block(IF_BLOCK);
  dim3 grid((n4 + IF_BLOCK - 1) / IF_BLOCK);  // 256 blocks = 2048 waves
  if_soft_reset_kernel<<<grid, block, 0, stream>>>((const f4*)cur, (f4*)out, n4);
}